// QA_Layer_partA_8315056685209
// MI455X (gfx1250) — compile-verified
//
#include <hip/hip_runtime.h>
#include <math.h>

#define B_ 32
#define C_ 512
#define Q_ 128
#define D_ 768
#define NEGBIG (-1e30f)

typedef float v2f __attribute__((ext_vector_type(2)));
typedef float v8f __attribute__((ext_vector_type(8)));

__device__ __forceinline__ v8f wmma_f32_k4(v2f a, v2f b, v8f c) {
    // D = A(16x4) * B(4x16) + C, all f32.  8-arg form:
    // (neg_a, A, neg_b, B, c_mod, C, reuse_a, reuse_b)
    return __builtin_amdgcn_wmma_f32_16x16x4_f32(false, a, false, b, (short)0, c, false, false);
}

// ---------------------------------------------------------------------------
// K1: rc[b,c] = cont[b,c,:]·w1 ; rq[b,q] = ques[b,q,:]·w2   (wave per row)
// ---------------------------------------------------------------------------
__global__ void k_rowdots(const float* __restrict__ cont, const float* __restrict__ ques,
                          const float* __restrict__ SW, float* __restrict__ rc,
                          float* __restrict__ rq) {
    const int t = threadIdx.x;
    const int gw = blockIdx.x * 8 + (t >> 5);
    const int lane = t & 31;
    const float* x; const float* w; float* dst;
    if (gw < B_ * C_) { x = cont + (size_t)gw * D_; w = SW;      dst = rc + gw; }
    else { int r = gw - B_ * C_; x = ques + (size_t)r * D_; w = SW + D_; dst = rq + r; }
    float acc = 0.f;
#pragma unroll
    for (int j = 0; j < 6; ++j) {
        const float4 xv = *reinterpret_cast<const float4*>(x + j * 128 + lane * 4);
        const float4 wv = *reinterpret_cast<const float4*>(w + j * 128 + lane * 4);
        acc += xv.x * wv.x + xv.y * wv.y + xv.z * wv.z + xv.w * wv.w;
    }
#pragma unroll
    for (int off = 16; off >= 1; off >>= 1) acc += __shfl_xor(acc, off, 32);
    if (lane == 0) *dst = acc;
}

// ---------------------------------------------------------------------------
// K0t: quesT[b][d][q] = ques[b][q][d]   (32x32 LDS tile transpose)
// ---------------------------------------------------------------------------
__global__ void k_transpose_q(const float* __restrict__ ques, float* __restrict__ quesT) {
    __shared__ float tile[32][33];
    const int b = blockIdx.z, q0 = blockIdx.y * 32, d0 = blockIdx.x * 32;
    const int tx = threadIdx.x, ty = threadIdx.y;
#pragma unroll
    for (int k = 0; k < 32; k += 8)
        tile[ty + k][tx] = ques[((size_t)(b * Q_ + q0 + ty + k)) * D_ + d0 + tx];
    __syncthreads();
#pragma unroll
    for (int k = 0; k < 32; k += 8)
        quesT[((size_t)(b * D_ + d0 + ty + k)) * Q_ + q0 + tx] = tile[tx][ty + k];
}

// ---------------------------------------------------------------------------
// K2: S[b,c,q] = rc + rq + (cont⊙w3)·ques   via V_WMMA_F32_16X16X4_F32
// block = 8 waves, macro-tile 32(c) x 128(q), K staged in LDS (chunks of 64)
// LDS row stride 68 dwords -> bank advance 4/row -> conflict-free b64 reads
// ---------------------------------------------------------------------------
__global__ void k_similarity(const float* __restrict__ cont, const float* __restrict__ ques,
                             const float* __restrict__ SW, const float* __restrict__ rc,
                             const float* __restrict__ rq, float* __restrict__ S) {
    __shared__ float As[32 * 68];
    __shared__ float Bs[128 * 68];
    const int b  = blockIdx.x >> 4;
    const int m0 = (blockIdx.x & 15) * 32;
    const int t = threadIdx.x;
    const int w = t >> 5, lane = t & 31;
    const int wm = w >> 2, wn = w & 3;
    const int n0 = wn * 32;
    const int h = lane >> 4, m = lane & 15;
    const float* w3 = SW + 2 * D_;

    v8f acc0 = {0, 0, 0, 0, 0, 0, 0, 0};
    v8f acc1 = {0, 0, 0, 0, 0, 0, 0, 0};

    for (int kc = 0; kc < D_; kc += 64) {
#pragma unroll
        for (int rep = 0; rep < 2; ++rep) {                 // stage A = (cont⊙w3) 32x64
            const int idx = t + rep * 256;
            const int row = idx >> 4, c4 = (idx & 15) * 4;
            float4 v = *reinterpret_cast<const float4*>(
                cont + ((size_t)(b * C_ + m0 + row)) * D_ + kc + c4);
            const float4 wv = *reinterpret_cast<const float4*>(w3 + kc + c4);
            v.x *= wv.x; v.y *= wv.y; v.z *= wv.z; v.w *= wv.w;
            *reinterpret_cast<float4*>(&As[row * 68 + c4]) = v;
        }
#pragma unroll
        for (int rep = 0; rep < 8; ++rep) {                 // stage B = ques 128x64
            const int idx = t + rep * 256;
            const int row = idx >> 4, c4 = (idx & 15) * 4;
            *reinterpret_cast<float4*>(&Bs[row * 68 + c4]) =
                *reinterpret_cast<const float4*>(
                    ques + ((size_t)(b * Q_ + row)) * D_ + kc + c4);
        }
        __syncthreads();
#pragma unroll
        for (int k = 0; k < 64; k += 4) {
            const v2f a  = *reinterpret_cast<const v2f*>(&As[(wm * 16 + m) * 68 + k + 2 * h]);
            const v2f b0 = *reinterpret_cast<const v2f*>(&Bs[(n0 + m) * 68 + k + 2 * h]);
            const v2f b1 = *reinterpret_cast<const v2f*>(&Bs[(n0 + 16 + m) * 68 + k + 2 * h]);
            acc0 = wmma_f32_k4(a, b0, acc0);
            acc1 = wmma_f32_k4(a, b1, acc1);
        }
        __syncthreads();
    }

    const float rq0 = rq[b * Q_ + n0 + m];
    const float rq1 = rq[b * Q_ + n0 + 16 + m];
#pragma unroll
    for (int i = 0; i < 8; ++i) {
        const int row = m0 + wm * 16 + i + 8 * h;
        const float rcv = rc[b * C_ + row];
        const size_t base = ((size_t)(b * C_ + row)) * Q_;
        S[base + n0 + m]      = acc0[i] + rcv + rq0;
        S[base + n0 + 16 + m] = acc1[i] + rcv + rq1;
    }
}

// ---------------------------------------------------------------------------
// K3a: per (b,q): colmax = max_c S, colinv = 1/sum_c exp(S-colmax)
// (ques_masks term is constant along axis C -> cancels in softmax exactly)
// ---------------------------------------------------------------------------
__global__ void k_colstats(const float* __restrict__ S, float* __restrict__ colmax,
                           float* __restrict__ colinv) {
    __shared__ float Lm[8][32], Ls[8][32];
    const int b = blockIdx.x >> 2;
    const int q0 = (blockIdx.x & 3) * 32;
    const int t = threadIdx.x, w = t >> 5, lane = t & 31;
    const int q = q0 + lane;
    float m = -INFINITY, s = 0.f;
    for (int i = 0; i < 64; ++i) {
        const int c = w + i * 8;
        const float v = S[((size_t)(b * C_ + c)) * Q_ + q];
        const float nm = fmaxf(m, v);
        s = s * __expf(m - nm) + __expf(v - nm);
        m = nm;
    }
    Lm[w][lane] = m; Ls[w][lane] = s;
    __syncthreads();
    if (w == 0) {
        float M = Lm[0][lane], Ssum = Ls[0][lane];
#pragma unroll
        for (int ww = 1; ww < 8; ++ww) {
            const float m2 = Lm[ww][lane], s2 = Ls[ww][lane];
            const float nm = fmaxf(M, m2);
            Ssum = Ssum * __expf(M - nm) + s2 * __expf(m2 - nm);
            M = nm;
        }
        colmax[b * Q_ + q] = M;
        colinv[b * Q_ + q] = 1.f / Ssum;
    }
}

// ---------------------------------------------------------------------------
// K3bc: S_max over q, cont-mask, softmax over C -> dist[b,c]  (block per b)
// ---------------------------------------------------------------------------
__global__ void k_cdist(const float* __restrict__ S, const int* __restrict__ cmask,
                        float* __restrict__ dist) {
    __shared__ float smax[C_];
    __shared__ float red[256];
    const int b = blockIdx.x;
    const int t = threadIdx.x, w = t >> 5, lane = t & 31;
    for (int it = 0; it < 64; ++it) {
        const int c = it * 8 + w;
        const float4 v = *reinterpret_cast<const float4*>(
            S + ((size_t)(b * C_ + c)) * Q_ + lane * 4);
        float m = fmaxf(fmaxf(v.x, v.y), fmaxf(v.z, v.w));
#pragma unroll
        for (int off = 16; off >= 1; off >>= 1) m = fmaxf(m, __shfl_xor(m, off, 32));
        if (lane == 0) smax[c] = m + (cmask[b * C_ + c] ? 0.f : NEGBIG);
    }
    __syncthreads();
    const float v0 = smax[t], v1 = smax[t + 256];
    red[t] = fmaxf(v0, v1);
    __syncthreads();
    for (int sfl = 128; sfl > 0; sfl >>= 1) {
        if (t < sfl) red[t] = fmaxf(red[t], red[t + sfl]);
        __syncthreads();
    }
    const float M = red[0];
    __syncthreads();
    const float e0 = __expf(v0 - M), e1 = __expf(v1 - M);
    red[t] = e0 + e1;
    __syncthreads();
    for (int sfl = 128; sfl > 0; sfl >>= 1) {
        if (t < sfl) red[t] += red[t + sfl];
        __syncthreads();
    }
    const float inv = 1.f / red[0];
    dist[b * C_ + t]       = e0 * inv;
    dist[b * C_ + t + 256] = e1 * inv;
}

// ---------------------------------------------------------------------------
// K3d: c_dash[b,d] = sum_c dist[b,c] * cont[b,c,d]
// ---------------------------------------------------------------------------
__global__ void k_cdash(const float* __restrict__ cont, const float* __restrict__ dist,
                        float* __restrict__ cdash) {
    const int b = blockIdx.x / 3;
    const int d = (blockIdx.x % 3) * 256 + threadIdx.x;
    float acc = 0.f;
    for (int c = 0; c < C_; ++c)
        acc += dist[b * C_ + c] * cont[((size_t)(b * C_ + c)) * D_ + d];
    cdash[b * D_ + d] = acc;
}

// ---------------------------------------------------------------------------
// K4: P = softmax(S, axis=C) recomputed into LDS; c2q tile = P @ quesT^T via
// WMMA; fused epilogue writes [c | c2q | c*c2q | c*c_dash] (the 192MiB pass)
// ---------------------------------------------------------------------------
__global__ void k_output(const float* __restrict__ S, const float* __restrict__ quesT,
                         const float* __restrict__ cont, const float* __restrict__ colmax,
                         const float* __restrict__ colinv, const float* __restrict__ cdash,
                         float* __restrict__ out) {
    __shared__ float Ps[16 * 132];   // stride 132 -> bank advance 4/row, conflict-free
    __shared__ float Cm[Q_], Ci[Q_];
    const int dblk = blockIdx.x * 128;
    const int c0   = blockIdx.y * 16;
    const int b    = blockIdx.z;
    const int t = threadIdx.x, w = t >> 5, lane = t & 31;
    const int h = lane >> 4, m = lane & 15;

    if (t < Q_) { Cm[t] = colmax[b * Q_ + t]; Ci[t] = colinv[b * Q_ + t]; }
    __syncthreads();
#pragma unroll
    for (int rep = 0; rep < 2; ++rep) {
        const int idx = t + rep * 256;
        const int row = idx >> 5, q4 = (idx & 31) * 4;
        const float4 v = *reinterpret_cast<const float4*>(
            S + ((size_t)(b * C_ + c0 + row)) * Q_ + q4);
        float4 p;
        p.x = __expf(v.x - Cm[q4 + 0]) * Ci[q4 + 0];
        p.y = __expf(v.y - Cm[q4 + 1]) * Ci[q4 + 1];
        p.z = __expf(v.z - Cm[q4 + 2]) * Ci[q4 + 2];
        p.w = __expf(v.w - Cm[q4 + 3]) * Ci[q4 + 3];
        *reinterpret_cast<float4*>(&Ps[row * 132 + q4]) = p;
    }
    __syncthreads();

    const int dcol = dblk + w * 16 + m;
    v8f acc = {0, 0, 0, 0, 0, 0, 0, 0};
    const float* qrow = quesT + ((size_t)(b * D_ + dcol)) * Q_;
#pragma unroll
    for (int k = 0; k < Q_; k += 4) {
        const v2f a  = *reinterpret_cast<const v2f*>(&Ps[m * 132 + k + 2 * h]);
        const v2f bq = *reinterpret_cast<const v2f*>(qrow + k + 2 * h);
        acc = wmma_f32_k4(a, bq, acc);
    }

    const float cd = cdash[b * D_ + dcol];
#pragma unroll
    for (int i = 0; i < 8; ++i) {
        const int crow = c0 + i + 8 * h;
        const float cv = cont[((size_t)(b * C_ + crow)) * D_ + dcol];
        const size_t ob = ((size_t)(b * C_ + crow)) * 3072 + dcol;
        const float v = acc[i];
        out[ob]        = cv;        // c
        out[ob + 768]  = v;         // c2q
        out[ob + 1536] = cv * v;    // c * c2q
        out[ob + 2304] = cv * cd;   // c * c_dash
    }
}

// ---------------------------------------------------------------------------
extern "C" void kernel_launch(void* const* d_in, const int* in_sizes, int n_in,
                              void* d_out, int out_size, void* d_ws, size_t ws_size,
                              hipStream_t stream) {
    const float* cont  = (const float*)d_in[0];
    const int*   cmask = (const int*)  d_in[1];
    const float* ques  = (const float*)d_in[2];
    // d_in[3] (ques_masks): constant along softmax axis C -> exact no-op
    const float* SW    = (const float*)d_in[4];
    float* out = (float*)d_out;

    float* ws     = (float*)d_ws;
    float* S      = ws;                                   // B*C*Q   = 2,097,152
    float* quesT  = S + (size_t)B_ * C_ * Q_;             // B*D*Q   = 3,145,728
    float* rc     = quesT + (size_t)B_ * D_ * Q_;         // B*C
    float* rq     = rc + B_ * C_;                         // B*Q
    float* colmax = rq + B_ * Q_;                         // B*Q
    float* colinv = colmax + B_ * Q_;                     // B*Q
    float* dist   = colinv + B_ * Q_;                     // B*C
    float* cdash  = dist + B_ * C_;                       // B*D

    k_rowdots   <<<2560, 256, 0, stream>>>(cont, ques, SW, rc, rq);
    k_transpose_q<<<dim3(24, 4, 32), dim3(32, 8), 0, stream>>>(ques, quesT);
    k_similarity<<<512, 256, 0, stream>>>(cont, ques, SW, rc, rq, S);
    k_colstats  <<<128, 256, 0, stream>>>(S, colmax, colinv);
    k_cdist     <<<32, 256, 0, stream>>>(S, cmask, dist);
    k_cdash     <<<96, 256, 0, stream>>>(cont, dist, cdash);
    k_output    <<<dim3(6, 32, 32), 256, 0, stream>>>(S, quesT, cont, colmax, colinv,
                                                      cdash, out);
}